// HyperCond_66236985639121
// MI455X (gfx1250) — compile-verified
//
#include <hip/hip_runtime.h>

// ---------------------------------------------------------------------------
// Problem constants (from the reference)
// ---------------------------------------------------------------------------
#define SBATCH   16
#define NINP     256
#define LCHUNK   16384
#define FMAPS    256
#define EMB_DIM  128
#define KWIDTH   3
#define NW       (FMAPS * KWIDTH)   // 768 rows of Ww
#define NTOT     (NW + FMAPS)       // 1024 total output columns (w ++ b)

typedef __attribute__((ext_vector_type(2))) float v2f;
typedef __attribute__((ext_vector_type(4))) float v4f;
typedef __attribute__((ext_vector_type(8))) float v8f;

// ---------------------------------------------------------------------------
// Kernel 1: hypernetwork GEMM via V_WMMA_F32_16X16X4_F32.
//   wcat[16 x 1024] = emb[16x128] @ [Ww;Wb]^T + [bw;bb]
// One wave (32 lanes) per 16x16 output tile; 64 tiles total; K-loop = 128/4.
//
// ISA layouts (CDNA5 ISA):
//   A 16x4 f32 : v0 = A[m, K=0] (lanes 0-15) / A[m, K=2] (lanes 16-31)
//                v1 = A[m, K=1]              / A[m, K=3]
//   B 4x16 f32 : v0 = B[K=0, n] (lanes 0-15) / B[K=2, n] (lanes 16-31)
//                v1 = B[K=1, n]              / B[K=3, n]
//   C/D 16x16  : lane l, vgpr r -> (M = (l<16 ? r : r+8), N = l&15)
// ---------------------------------------------------------------------------
__global__ __launch_bounds__(32) void hyper_gemm_wmma(
    const float* __restrict__ emb, const float* __restrict__ Ww,
    const float* __restrict__ bw,  const float* __restrict__ Wb,
    const float* __restrict__ bb,  float* __restrict__ wcat)
{
    const int lane   = threadIdx.x;      // 0..31
    const int half   = lane >> 4;        // 0: K lo pair, 1: K hi pair
    const int mn     = lane & 15;        // A row index (M) and B col index (N)
    const int nbase  = blockIdx.x * 16;  // output-column tile base
    const int j      = nbase + mn;       // output column this lane feeds into B

    // B row source: concatenated [Ww ; Wb]; nbase is a multiple of 16 and
    // NW=768 is too, so this is uniform per block.
    const float* __restrict__ wrow =
        (j < NW) ? (Ww + (size_t)j * EMB_DIM) : (Wb + (size_t)(j - NW) * EMB_DIM);
    const float bias = (j < NW) ? bw[j] : bb[j - NW];

    const float* __restrict__ arow = emb + (size_t)mn * EMB_DIM;

    // Bias depends only on N -> preload into every accumulator VGPR.
    v8f c;
#pragma unroll
    for (int r = 0; r < 8; ++r) c[r] = bias;

#pragma unroll 4
    for (int kb = 0; kb < EMB_DIM; kb += 4) {
        const int kk = kb + 2 * half;
        v2f a; a.x = arow[kk]; a.y = arow[kk + 1];
        v2f b; b.x = wrow[kk]; b.y = wrow[kk + 1];
        // 8 args: (neg_a, A, neg_b, B, c_mod, C, reuse_a, reuse_b)
        c = __builtin_amdgcn_wmma_f32_16x16x4_f32(
                false, a, false, b, (short)0, c, false, false);
    }

#pragma unroll
    for (int r = 0; r < 8; ++r) {
        const int M = half ? (r + 8) : r;
        wcat[(size_t)M * NTOT + nbase + mn] = c[r];
    }
}

// ---------------------------------------------------------------------------
// Kernel 2: streaming depthwise width-3 conv (bandwidth-bound, ~512 MB).
//   out[s,f,l] = b + w0*h[l-1] + w1*h[l] + w2*h[l+1]   (zero-padded ends)
// b128 loads, FMA, non-temporal b128 stores (output never re-read; keep the
// 192MB L2 for h reuse instead).
// ---------------------------------------------------------------------------
__global__ __launch_bounds__(256) void hyper_conv(
    const float* __restrict__ h, const float* __restrict__ wcat,
    float* __restrict__ out)
{
    const int s = blockIdx.z;
    const int f = blockIdx.y;
    const int l = (blockIdx.x * 256 + threadIdx.x) * 4;   // 4 floats / thread

    // Per-(s,f) scalars — uniform across the block -> scalar loads.
    const float* __restrict__ wrow = wcat + (size_t)s * NTOT;
    const float w0 = wrow[f * 3 + 0];
    const float w1 = wrow[f * 3 + 1];
    const float w2 = wrow[f * 3 + 2];
    const float bb = wrow[NW + f];

    const size_t row = ((size_t)s * NINP + f) * LCHUNK;
    const float* __restrict__ hr = h + row;

    const v4f cen = *(const v4f*)(hr + l);
    const float left  = (l > 0)            ? hr[l - 1] : 0.0f;  // pad
    const float right = (l + 4 < LCHUNK)   ? hr[l + 4] : 0.0f;  // pad

    v4f o;
    o.x = __builtin_fmaf(w0, left,  __builtin_fmaf(w1, cen.x, __builtin_fmaf(w2, cen.y, bb)));
    o.y = __builtin_fmaf(w0, cen.x, __builtin_fmaf(w1, cen.y, __builtin_fmaf(w2, cen.z, bb)));
    o.z = __builtin_fmaf(w0, cen.y, __builtin_fmaf(w1, cen.z, __builtin_fmaf(w2, cen.w, bb)));
    o.w = __builtin_fmaf(w0, cen.z, __builtin_fmaf(w1, cen.w, __builtin_fmaf(w2, right, bb)));

    __builtin_nontemporal_store(o, (v4f*)(out + row + l));
}

// ---------------------------------------------------------------------------
// Harness entry point
// ---------------------------------------------------------------------------
extern "C" void kernel_launch(void* const* d_in, const int* in_sizes, int n_in,
                              void* d_out, int out_size, void* d_ws, size_t ws_size,
                              hipStream_t stream) {
    (void)in_sizes; (void)n_in; (void)out_size; (void)ws_size;

    const float* h   = (const float*)d_in[0];  // (16, 256, 16384)
    const float* emb = (const float*)d_in[1];  // (16, 128)
    const float* Ww  = (const float*)d_in[2];  // (768, 128)
    const float* bw  = (const float*)d_in[3];  // (768,)
    const float* Wb  = (const float*)d_in[4];  // (256, 128)
    const float* bb  = (const float*)d_in[5];  // (256,)
    float* out  = (float*)d_out;               // (16, 256, 16384)
    float* wcat = (float*)d_ws;                // 16*1024 floats = 64 KB scratch

    // 1) 16x1024x128 hypernetwork GEMM (+bias) -> wcat, one wave per tile.
    hyper_gemm_wmma<<<dim3(NTOT / 16), dim3(32), 0, stream>>>(
        emb, Ww, bw, Wb, bb, wcat);

    // 2) Streaming conv: grid (L/1024, FMAPS, SBATCH), 256 threads, 4 f32/thr.
    dim3 grid(LCHUNK / (256 * 4), FMAPS, SBATCH);
    hyper_conv<<<grid, dim3(256), 0, stream>>>(h, wcat, out);
}